// DenseDilatedKnnGraphDGL_420906795278
// MI455X (gfx1250) — compile-verified
//
#include <hip/hip_runtime.h>
#include <hip/hip_bf16.h>

#define B_ 8
#define C_ 64
#define N_ 4096
#define K_ 9
#define MT 16            // rows per workgroup
#define CHUNK 512        // columns of distances buffered in LDS per pass
#define DSTRIDE 516      // padded LDS row stride (floats) for bank spread
#define NCHUNK (N_ / CHUNK)
#define NWAVE 8

typedef __attribute__((ext_vector_type(2))) float v2f;
typedef __attribute__((ext_vector_type(8))) float v8f;

// Kernel 1: per-point squared norms sq[b][n] = sum_c x[b][c][n]^2
__global__ __launch_bounds__(256) void knn_sq_kernel(const float* __restrict__ x,
                                                     float* __restrict__ sq) {
  int t = blockIdx.x * blockDim.x + threadIdx.x;   // 0 .. B*N-1
  int b = t >> 12;                                  // N_ = 4096
  int n = t & (N_ - 1);
  const float* xb = x + (size_t)b * C_ * N_ + n;
  float s = 0.f;
#pragma unroll
  for (int c = 0; c < C_; ++c) {
    float v = xb[(size_t)c * N_];
    s += v * v;
  }
  sq[t] = s;
}

// Kernel 2: one workgroup = (batch b, 16-row tile). 8 waves compute distance
// tiles with V_WMMA_F32_16X16X4_F32, stage chunks of 512 columns in LDS,
// 16 threads/row maintain running top-9, leader merges and writes outputs.
__global__ __launch_bounds__(256) void knn_topk_kernel(const float* __restrict__ x,
                                                       const float* __restrict__ sq,
                                                       int* __restrict__ srcO,
                                                       int* __restrict__ dstO,
                                                       float* __restrict__ dO) {
  __shared__ float lds[MT * DSTRIDE];   // 33,024 B

  const int wg = blockIdx.x;
  const int b = wg >> 8;                 // N_/MT = 256 row-tiles per batch
  const int r0 = (wg & 255) * MT;
  const int tid = threadIdx.x;
  const int w = tid >> 5;
  const int lane = tid & 31;
  const int h = lane >> 4;               // half-wave select (K pair / M+8)
  const int ln = lane & 15;

  const float* xb = x + (size_t)b * C_ * N_;
  const float* sqb = sq + b * N_;

  // A fragments: 16x4 f32 layout -> lanes 0-15 hold M=ln, VGPR pair holds
  // K = 2h, 2h+1 within each kb step of 4 channels. Loaded once, reused for
  // all 256 column tiles.
  v2f a[16];
#pragma unroll
  for (int kb = 0; kb < 16; ++kb) {
    int c = 4 * kb + 2 * h;
    a[kb].x = xb[(size_t)c * N_ + r0 + ln];
    a[kb].y = xb[(size_t)(c + 1) * N_ + r0 + ln];
  }
  // Row squared-norms: acc VGPR v holds M = v + 8h  ->  rq[v] = sq[r0+8h+v]
  float rq[8];
#pragma unroll
  for (int v = 0; v < 8; ++v) rq[v] = sqb[r0 + 8 * h + v];

  // Selection role: 16 threads per row, interleaved columns.
  const int srow = tid >> 4;   // 0..15
  const int sj = tid & 15;
  float td[K_];
  int ti[K_];
#pragma unroll
  for (int q = 0; q < K_; ++q) { td[q] = 3.0e38f; ti[q] = 0x7fffffff; }

  for (int ch = 0; ch < NCHUNK; ++ch) {
    const int cbase = ch * CHUNK;
    // ---- compute phase: each wave owns column tiles t = w, w+8, ... ----
    for (int t = w; t < CHUNK / 16; t += NWAVE) {
      const int col = cbase + t * 16 + ln;
      v8f acc = {0.f, 0.f, 0.f, 0.f, 0.f, 0.f, 0.f, 0.f};
#pragma unroll
      for (int kb = 0; kb < 16; ++kb) {
        int c = 4 * kb + 2 * h;
        v2f bv;
        bv.x = xb[(size_t)c * N_ + col];
        bv.y = xb[(size_t)(c + 1) * N_ + col];
        acc = __builtin_amdgcn_wmma_f32_16x16x4_f32(
            /*neg_a=*/false, a[kb], /*neg_b=*/false, bv,
            /*c_mod=*/(short)0, acc, /*reuse_a=*/false, /*reuse_b=*/false);
      }
      const float sc = sqb[col];
#pragma unroll
      for (int v = 0; v < 8; ++v) {
        // acc VGPR v: lanes 0-15 -> M=v, lanes 16-31 -> M=v+8; N = ln
        lds[(v + 8 * h) * DSTRIDE + t * 16 + ln] = rq[v] + sc - 2.0f * acc[v];
      }
    }
    __syncthreads();
    // ---- selection phase: running top-9 (lexicographic ties, JAX-style) ----
#pragma unroll 4
    for (int i = 0; i < CHUNK / 16; ++i) {
      const int cl = sj + (i << 4);
      const float d = lds[srow * DSTRIDE + cl];
      const int gi = cbase + cl;
      if ((d < td[K_ - 1]) || (d == td[K_ - 1] && gi < ti[K_ - 1])) {
        int p = K_ - 1;
        while (p > 0 && ((d < td[p - 1]) || (d == td[p - 1] && gi < ti[p - 1]))) {
          td[p] = td[p - 1];
          ti[p] = ti[p - 1];
          --p;
        }
        td[p] = d;
        ti[p] = gi;
      }
    }
    __syncthreads();
  }

  // ---- merge: 16 sorted 9-lists per row -> final top-9 (reuse LDS) ----
  float* mD = lds;                                // 2304 floats
  int* mI = (int*)(lds + MT * 16 * K_);           // 2304 ints, total 18,432 B
#pragma unroll
  for (int q = 0; q < K_; ++q) {
    mD[(srow * 16 + sj) * K_ + q] = td[q];
    mI[(srow * 16 + sj) * K_ + q] = ti[q];
  }
  __syncthreads();

  if (sj == 0) {
    int head[16];
#pragma unroll
    for (int l = 0; l < 16; ++l) head[l] = 0;
    const int node = r0 + srow;
    const long base = ((long)b * N_ + node) * K_;
    for (int q = 0; q < K_; ++q) {
      float bd = 3.39e38f;
      int bi = 0x7fffffff;
      int bl = 0;
#pragma unroll
      for (int l = 0; l < 16; ++l) {
        const int hh = head[l];
        const float d = (hh < K_) ? mD[(srow * 16 + l) * K_ + hh] : 3.4e38f;
        const int i = (hh < K_) ? mI[(srow * 16 + l) * K_ + hh] : 0x7fffffff;
        if ((d < bd) || (d == bd && i < bi)) { bd = d; bi = i; bl = l; }
      }
      head[bl]++;
      srcO[base + q] = bi + b * N_;
      dstO[base + q] = node + b * N_;
      dO[base + q] = bd;
    }
  }
}

extern "C" void kernel_launch(void* const* d_in, const int* in_sizes, int n_in,
                              void* d_out, int out_size, void* d_ws, size_t ws_size,
                              hipStream_t stream) {
  (void)in_sizes; (void)n_in; (void)out_size; (void)ws_size;
  const float* x = (const float*)d_in[0];   // (8, 64, 4096) f32
  // d_in[1] is k (== 9), baked in as K_.

  float* sqw = (float*)d_ws;                // B*N f32 = 128 KB scratch

  int* srcO = (int*)d_out;                  // (B*N*k,) int32
  int* dstO = srcO + B_ * N_ * K_;          // (B*N*k,) int32
  float* dO = (float*)d_out + 2 * B_ * N_ * K_;  // (B,N,k) f32

  knn_sq_kernel<<<(B_ * N_) / 256, 256, 0, stream>>>(x, sqw);
  knn_topk_kernel<<<B_ * (N_ / MT), 256, 0, stream>>>(x, sqw, srcO, dstO, dO);
}